// YoloNASPoseLoss_63874753626213
// MI455X (gfx1250) — compile-verified
//
#include <hip/hip_runtime.h>
#include <math.h>

// ---------------------------------------------------------------------------
// YOLO-NAS pose loss for MI455X (gfx1250, wave32).
// f32 end-to-end (loss computation). Memory-bound: ~250MB traffic -> ~11us
// roofline at 23.3 TB/s. WMMA (V_WMMA_F32_16X16X4_F32) used for the DFL
// softmax-expectation contraction; top-k runs out of LDS.
// ---------------------------------------------------------------------------

typedef float v2f __attribute__((ext_vector_type(2)));
typedef float v8f __attribute__((ext_vector_type(8)));

constexpr int   kB    = 16;
constexpr int   kL    = 8400;
constexpr int   kN    = 32;      // padded GTs per image
constexpr int   kJ    = 17;      // keypoints
constexpr int   kBins = 17;      // REG_MAX + 1
constexpr int   kTopK = 13;
constexpr float kEps  = 1e-9f;

// ---------------------------------------------------------------------------
// K1: DFL bbox decode. One wave32 per 4 anchors: 16 rows = (anchor,side).
// dist = sum_k softmax(x)_k * k computed as one 16x16x4 f32 WMMA chain:
//   A(16x4)=exp(x-max) chunks, B col0=1 (denominator), col1=bin index.
// EXEC is all ones at the WMMA (exact grid, no divergence before it).
// ---------------------------------------------------------------------------
__global__ __launch_bounds__(256) void ynp_decode(
    const float* __restrict__ pd, const float* __restrict__ ap,
    const float* __restrict__ st, float* __restrict__ pbs)
{
  const int lane = threadIdx.x & 31;
  const int grp  = blockIdx.x * (blockDim.x >> 5) + (threadIdx.x >> 5);
  const int b    = grp / (kL / 4);
  const int l4   = (grp - b * (kL / 4)) * 4;
  const int m    = lane & 15;      // row = (anchor within group, side)
  const int h    = lane >> 4;      // K-half: this lane holds K=2h,2h+1
  const int l    = l4 + (m >> 2);
  const int s    = m & 3;

  const float* base = pd + (size_t)(b * kL + l) * 68 + s * kBins;

  float xv[10];
  float mx = -3.0e38f;
#pragma unroll
  for (int c = 0; c < 5; ++c) {
    const int k0 = 4 * c + 2 * h;
    const float v0 = (k0     < kBins) ? base[k0]     : -3.0e38f;
    const float v1 = (k0 + 1 < kBins) ? base[k0 + 1] : -3.0e38f;
    xv[2 * c] = v0; xv[2 * c + 1] = v1;
    mx = fmaxf(mx, fmaxf(v0, v1));
  }
  mx = fmaxf(mx, __shfl_xor(mx, 16, 32));   // combine the two K-halves of a row

  v8f acc = {0.f, 0.f, 0.f, 0.f, 0.f, 0.f, 0.f, 0.f};
#pragma unroll
  for (int c = 0; c < 5; ++c) {
    const int k0 = 4 * c + 2 * h;
    v2f a;                       // A 16x4: lane m (h=0) K=0,1 ; lane m+16 K=2,3
    a.x = expf(xv[2 * c]     - mx);      // underflows to 0 for padded bins
    a.y = expf(xv[2 * c + 1] - mx);
    v2f w;                       // B 4x16: col 0 -> sum, col 1 -> dot(bin idx)
    w.x = (k0     < kBins) ? ((m == 0) ? 1.f : (m == 1) ? (float)k0       : 0.f) : 0.f;
    w.y = (k0 + 1 < kBins) ? ((m == 0) ? 1.f : (m == 1) ? (float)(k0 + 1) : 0.f) : 0.f;
    acc = __builtin_amdgcn_wmma_f32_16x16x4_f32(
        /*neg_a=*/false, a, /*neg_b=*/false, w,
        /*c_mod=*/(short)0, acc, /*reuse_a=*/false, /*reuse_b=*/false);
  }

  // Extract D[m][0] (sum) and D[m][1] (dot): VGPR r holds rows r (lanes 0-15)
  // and r+8 (lanes 16-31); row data sits in lane==col.
  const int src = (m & 8) ? 16 : 0;
  float sum_m = 1.f, dot_m = 0.f;
#pragma unroll
  for (int r = 0; r < 8; ++r) {
    const float vs = __shfl(acc[r], src,     32);
    const float vd = __shfl(acc[r], src + 1, 32);
    if ((m & 7) == r) { sum_m = vs; dot_m = vd; }
  }
  const float dist = dot_m / sum_m;

  const float stv = st[l];
  const float a0  = ap[2 * l]     / stv;
  const float a1  = ap[2 * l + 1] / stv;
  float val;
  if      (s == 0) val = a0 - dist;
  else if (s == 1) val = a1 - dist;
  else if (s == 2) val = a0 + dist;
  else             val = a1 + dist;
  if (h == 0) pbs[(size_t)(b * kL + l) * 4 + s] = val;
}

// ---------------------------------------------------------------------------
// K2: per-(b,g,l) alignment metric sig(score) * iou^6 (raw, no in-GT gate --
// the gate is applied in top-k; raw metric is needed after max-IoU override).
// Also clears mask_pos for this call (deterministic across graph replays).
// ---------------------------------------------------------------------------
__global__ __launch_bounds__(256) void ynp_metric(
    const float* __restrict__ ps, const float* __restrict__ pbs,
    const float* __restrict__ st, const float* __restrict__ gtb,
    float* __restrict__ metrics, unsigned char* __restrict__ mask)
{
  const int bg = blockIdx.y;
  const int b  = bg >> 5;
  const int l  = blockIdx.x * blockDim.x + threadIdx.x;
  if (l >= kL) return;

  const float* gb = gtb + (size_t)bg * 4;
  const float gx1 = gb[0], gy1 = gb[1], gx2 = gb[2], gy2 = gb[3];
  const float stv = st[l];
  const float* pb = pbs + (size_t)(b * kL + l) * 4;
  const float px1 = pb[0] * stv, py1 = pb[1] * stv;
  const float px2 = pb[2] * stv, py2 = pb[3] * stv;

  const float iw    = fmaxf(fminf(gx2, px2) - fmaxf(gx1, px1), 0.f);
  const float ih    = fmaxf(fminf(gy2, py2) - fmaxf(gy1, py1), 0.f);
  const float inter = iw * ih;
  const float ag    = (gx2 - gx1) * (gy2 - gy1);
  const float apA   = (px2 - px1) * (py2 - py1);
  const float iou   = inter / (ag + apA - inter + kEps);

  const float sc = ps[b * kL + l];
  const float sg = 1.f / (1.f + expf(-sc));
  const float i2 = iou * iou;
  metrics[(size_t)bg * kL + l] = sg * i2 * i2 * i2;
  mask[(size_t)bg * kL + l]    = 0;
}

// ---------------------------------------------------------------------------
// K3: top-13 per (b,g). Metrics row (gated by in-GT) staged in LDS (33.6 KB,
// well inside the 320 KB WGP budget); 13 argmax passes tombstone winners.
// Tie-break = lowest index (matches jax.lax.top_k). Fixed-order tree reduce.
// ---------------------------------------------------------------------------
__global__ __launch_bounds__(256) void ynp_topk(
    const float* __restrict__ metrics, const float* __restrict__ ap,
    const float* __restrict__ gtb, const float* __restrict__ padm,
    unsigned char* __restrict__ mask)
{
  const int bg  = blockIdx.x;
  const int tid = threadIdx.x;
  __shared__ float smet[kL];
  __shared__ float sv[256];
  __shared__ int   si[256];
  __shared__ int   chosen[kTopK];

  const float* gb = gtb + (size_t)bg * 4;
  const float gx1 = gb[0], gy1 = gb[1], gx2 = gb[2], gy2 = gb[3];
  const bool  pad = padm[bg] > 0.f;
  const float* mrow = metrics + (size_t)bg * kL;

  for (int l = tid; l < kL; l += 256) {
    const float axp = ap[2 * l], ayp = ap[2 * l + 1];
    const bool ing = (axp - gx1 > kEps) && (ayp - gy1 > kEps) &&
                     (gx2 - axp > kEps) && (gy2 - ayp > kEps);
    smet[l] = ing ? mrow[l] : 0.f;
  }
  __syncthreads();

  for (int k = 0; k < kTopK; ++k) {
    float bv = -0.5f;           // all live entries are >= 0, tombstones are -1
    int   bi = 0x3fffffff;
    for (int l = tid; l < kL; l += 256) {
      const float v = smet[l];
      if (v > bv || (v == bv && l < bi)) { bv = v; bi = l; }
    }
    sv[tid] = bv; si[tid] = bi;
    __syncthreads();
    for (int off = 128; off > 0; off >>= 1) {
      if (tid < off) {
        if (sv[tid + off] > sv[tid] ||
            (sv[tid + off] == sv[tid] && si[tid + off] < si[tid])) {
          sv[tid] = sv[tid + off]; si[tid] = si[tid + off];
        }
      }
      __syncthreads();
    }
    if (tid == 0) { chosen[k] = si[0]; smet[si[0]] = -1.f; }
    __syncthreads();
  }

  if (tid < kTopK) {
    const int l = chosen[tid];
    const float axp = ap[2 * l], ayp = ap[2 * l + 1];
    const bool ing = (axp - gx1 > kEps) && (ayp - gy1 > kEps) &&
                     (gx2 - axp > kEps) && (gy2 - ayp > kEps);
    mask[(size_t)bg * kL + l] = (ing && pad) ? 1 : 0;
  }
}

// ---------------------------------------------------------------------------
// K4: per-anchor assignment. Multi-match anchors replaced by max-IoU one-hot
// (argmax over all 32 GTs, lowest index on tie). Writes final mask column,
// pos flag and assigned GT index.
// ---------------------------------------------------------------------------
__global__ __launch_bounds__(256) void ynp_assign(
    const float* __restrict__ pbs, const float* __restrict__ st,
    const float* __restrict__ gtb, unsigned char* __restrict__ mask,
    int* __restrict__ agi, unsigned char* __restrict__ pos)
{
  const int i = blockIdx.x * blockDim.x + threadIdx.x;
  if (i >= kB * kL) return;
  const int b = i / kL, l = i - b * kL;

  unsigned int bits = 0;
  for (int g = 0; g < kN; ++g)
    if (mask[((size_t)(b * kN + g)) * kL + l]) bits |= (1u << g);
  int cnt = __popc(bits);

  if (cnt > 1) {
    const float stv = st[l];
    const float* pb = pbs + (size_t)i * 4;
    const float px1 = pb[0] * stv, py1 = pb[1] * stv;
    const float px2 = pb[2] * stv, py2 = pb[3] * stv;
    const float apA = (px2 - px1) * (py2 - py1);
    float bestv = -1.f; int bestg = 0;
    for (int g = 0; g < kN; ++g) {
      const float* gb = gtb + (size_t)(b * kN + g) * 4;
      const float gx1 = gb[0], gy1 = gb[1], gx2 = gb[2], gy2 = gb[3];
      const float iw    = fmaxf(fminf(gx2, px2) - fmaxf(gx1, px1), 0.f);
      const float ih    = fmaxf(fminf(gy2, py2) - fmaxf(gy1, py1), 0.f);
      const float inter = iw * ih;
      const float ag    = (gx2 - gx1) * (gy2 - gy1);
      const float iou   = inter / (ag + apA - inter + kEps);
      if (iou > bestv) { bestv = iou; bestg = g; }
    }
    bits = (1u << bestg);
    for (int g = 0; g < kN; ++g)
      mask[((size_t)(b * kN + g)) * kL + l] = (g == bestg) ? 1 : 0;
    cnt = 1;
  }
  agi[i] = bits ? (__ffs(bits) - 1) : 0;
  pos[i] = (cnt > 0) ? 1 : 0;
}

// ---------------------------------------------------------------------------
// K5: per-(b,g) maxima over L of metric*mask and iou*mask (final mask).
// ---------------------------------------------------------------------------
__global__ __launch_bounds__(256) void ynp_gtmax(
    const float* __restrict__ metrics, const unsigned char* __restrict__ mask,
    const float* __restrict__ pbs, const float* __restrict__ st,
    const float* __restrict__ gtb, float* __restrict__ mmax,
    float* __restrict__ imax)
{
  const int bg  = blockIdx.x;
  const int b   = bg >> 5;
  const int tid = threadIdx.x;
  __shared__ float sm[256], sio[256];

  const float* gb = gtb + (size_t)bg * 4;
  const float gx1 = gb[0], gy1 = gb[1], gx2 = gb[2], gy2 = gb[3];
  const float ag  = (gx2 - gx1) * (gy2 - gy1);

  float lm = 0.f, li = 0.f;
  for (int l = tid; l < kL; l += 256) {
    if (!mask[(size_t)bg * kL + l]) continue;
    lm = fmaxf(lm, metrics[(size_t)bg * kL + l]);
    const float stv = st[l];
    const float* pb = pbs + (size_t)(b * kL + l) * 4;
    const float px1 = pb[0] * stv, py1 = pb[1] * stv;
    const float px2 = pb[2] * stv, py2 = pb[3] * stv;
    const float iw    = fmaxf(fminf(gx2, px2) - fmaxf(gx1, px1), 0.f);
    const float ih    = fmaxf(fminf(gy2, py2) - fmaxf(gy1, py1), 0.f);
    const float inter = iw * ih;
    const float apA   = (px2 - px1) * (py2 - py1);
    li = fmaxf(li, inter / (ag + apA - inter + kEps));
  }
  sm[tid] = lm; sio[tid] = li;
  __syncthreads();
  for (int off = 128; off > 0; off >>= 1) {
    if (tid < off) {
      sm[tid]  = fmaxf(sm[tid],  sm[tid + off]);
      sio[tid] = fmaxf(sio[tid], sio[tid + off]);
    }
    __syncthreads();
  }
  if (tid == 0) { mmax[bg] = sm[0]; imax[bg] = sio[0]; }
}

// ---------------------------------------------------------------------------
// K7: fused per-anchor losses. 8 partial sums per block, fixed-order LDS tree
// reduction -> deterministic. Partials: [assigned_scores, cls, iou, dfl,
// pos, kmf, pose_reg, pose_cls].
// ---------------------------------------------------------------------------
__global__ __launch_bounds__(256) void ynp_loss(
    const float* __restrict__ ps,  const float* __restrict__ pd,
    const float* __restrict__ ppl, const float* __restrict__ ap,
    const float* __restrict__ st,  const float* __restrict__ gtb,
    const float* __restrict__ gposes, const float* __restrict__ sigs,
    const float* __restrict__ pbs, const float* __restrict__ metrics,
    const unsigned char* __restrict__ mask, const int* __restrict__ agi,
    const unsigned char* __restrict__ pos, const float* __restrict__ mmax,
    const float* __restrict__ imax, float* __restrict__ partials)
{
  const int tid = threadIdx.x;
  const int i   = blockIdx.x * blockDim.x + tid;
  float acc[8] = {0.f, 0.f, 0.f, 0.f, 0.f, 0.f, 0.f, 0.f};

  if (i < kB * kL) {
    const int b = i / kL, l = i - b * kL;
    const int bgBase = b * kN;

    // assigned score: max_g( metric / (mmax+eps) * imax ) over final mask
    float as = 0.f;
    for (int g = 0; g < kN; ++g) {
      if (mask[((size_t)(bgBase + g)) * kL + l]) {
        const float mv = metrics[((size_t)(bgBase + g)) * kL + l];
        as = fmaxf(as, mv / (mmax[bgBase + g] + kEps) * imax[bgBase + g]);
      }
    }
    const float posf = pos[i] ? 1.f : 0.f;

    // classification loss (varifocal-style weighting)
    const float sc  = ps[i];
    const float sg  = 1.f / (1.f + expf(-sc));
    const float w   = (sg - as) * (sg - as);
    const float bce = fmaxf(sc, 0.f) - sc * as + log1pf(expf(-fabsf(sc)));
    acc[0] = as;
    acc[1] = w * bce;

    const float stv  = st[l];
    const float aps0 = ap[2 * l] / stv, aps1 = ap[2 * l + 1] / stv;
    const int   g    = agi[i];
    const float* gb  = gtb + (size_t)(bgBase + g) * 4;
    const float ab0 = gb[0] / stv, ab1 = gb[1] / stv;
    const float ab2 = gb[2] / stv, ab3 = gb[3] / stv;
    const float* pb = pbs + (size_t)i * 4;
    const float pb0 = pb[0], pb1 = pb[1], pb2 = pb[2], pb3 = pb[3];

    // GIoU
    const float iw    = fmaxf(fminf(pb2, ab2) - fmaxf(pb0, ab0), 0.f);
    const float ih    = fmaxf(fminf(pb3, ab3) - fmaxf(pb1, ab1), 0.f);
    const float inter = iw * ih;
    const float a1a   = (pb2 - pb0) * (pb3 - pb1);
    const float a2a   = (ab2 - ab0) * (ab3 - ab1);
    const float uni   = a1a + a2a - inter + kEps;
    const float iou   = inter / uni;
    const float cw    = fmaxf(pb2, ab2) - fminf(pb0, ab0);
    const float ch    = fmaxf(pb3, ab3) - fminf(pb1, ab1);
    const float areac = cw * ch + kEps;
    const float gl    = 1.f - (iou - (areac - uni) / areac);
    const float bw    = as * posf;
    acc[2] = gl * bw;

    // DFL (recompute log-softmax per side; L2-hot re-reads)
    const float tg[4] = { aps0 - ab0, aps1 - ab1, ab2 - aps0, ab3 - aps1 };
    const float* prow = pd + (size_t)i * 68;
    float dsum = 0.f;
#pragma unroll
    for (int s = 0; s < 4; ++s) {
      const float* bp = prow + s * kBins;
      float mx = bp[0];
      for (int k = 1; k < kBins; ++k) mx = fmaxf(mx, bp[k]);
      float se = 0.f;
      for (int k = 0; k < kBins; ++k) se += expf(bp[k] - mx);
      const float lse = mx + logf(se);
      const float t   = fminf(fmaxf(tg[s], 0.f), 15.99f);   // REG_MAX - 0.01
      const int   tl  = (int)floorf(t);
      const float wl  = (float)tl + 1.f - t;
      const float ll  = lse - bp[tl];
      const float lr  = lse - bp[tl + 1];
      dsum += ll * wl + lr * (1.f - wl);
    }
    acc[3] = dsum * 0.25f * bw;
    acc[4] = posf;

    // pose (OKS-style reg + visibility cls)
    const float area = (ab2 - ab0) * (ab3 - ab1);
    const float* kpb = gposes + (size_t)(bgBase + g) * kJ * 3;
    const float* plb = ppl + (size_t)i * kJ * 3;
    __builtin_prefetch(plb, 0, 3);   // -> global_prefetch_b8
    float kc = 0.f, kr = 0.f, kcl = 0.f;
#pragma unroll
    for (int j = 0; j < kJ; ++j) {
      const float gx  = kpb[3 * j]     / stv;
      const float gy  = kpb[3 * j + 1] / stv;
      const float vis = kpb[3 * j + 2];
      const float px  = plb[3 * j]     + aps0;
      const float py  = plb[3 * j + 1] + aps1;
      const float lv  = plb[3 * j + 2];
      const float dx = px - gx, dy = py - gy;
      const float d  = dx * dx + dy * dy;
      const float km = (vis != 0.f && pos[i]) ? 1.f : 0.f;
      const float sj = sigs[j];
      const float e  = d / (4.f * sj * sj) / (area + kEps) * 0.5f;
      kr  += (1.f - expf(-e)) * km;
      kc  += km;
      kcl += (fmaxf(lv, 0.f) - lv * km + log1pf(expf(-fabsf(lv)))) * posf;
    }
    acc[5] = kc; acc[6] = kr; acc[7] = kcl;
  }

  __shared__ float red[256];
#pragma unroll
  for (int jcomp = 0; jcomp < 8; ++jcomp) {
    red[tid] = acc[jcomp];
    __syncthreads();
    for (int off = 128; off > 0; off >>= 1) {
      if (tid < off) red[tid] += red[tid + off];
      __syncthreads();
    }
    if (tid == 0) partials[(size_t)blockIdx.x * 8 + jcomp] = red[0];
    __syncthreads();
  }
}

// ---------------------------------------------------------------------------
// K8: serial deterministic finalize. d_out = [loss, cls, iou, dfl, pose_cls,
// pose_reg, loss] (tuple (loss, log_losses) flattened).
// ---------------------------------------------------------------------------
__global__ void ynp_final(const float* __restrict__ partials, int nb,
                          float* __restrict__ out)
{
  if (threadIdx.x != 0 || blockIdx.x != 0) return;
  float S[8] = {0.f, 0.f, 0.f, 0.f, 0.f, 0.f, 0.f, 0.f};
  for (int p = 0; p < nb; ++p)
    for (int j = 0; j < 8; ++j) S[j] += partials[(size_t)p * 8 + j];
  const float ass = fmaxf(S[0], 1.f);
  const float lc  = S[1] / ass;
  const float li  = S[2] / ass;
  const float ld  = S[3] / ass;
  const float npk = fmaxf(S[4] * (float)kJ, 1.f);
  const float factor = npk / (S[5] + kEps);
  const float lpr = factor * S[6] / npk;
  const float lpc = S[7] / npk;
  const float loss = lc + 2.5f * li + 0.5f * ld + lpc + lpr;
  out[0] = loss; out[1] = lc; out[2] = li; out[3] = ld;
  out[4] = lpc;  out[5] = lpr; out[6] = loss;
}

// ---------------------------------------------------------------------------
extern "C" void kernel_launch(void* const* d_in, const int* in_sizes, int n_in,
                              void* d_out, int out_size, void* d_ws,
                              size_t ws_size, hipStream_t stream)
{
  (void)in_sizes; (void)n_in; (void)out_size; (void)ws_size;
  const float* ps     = (const float*)d_in[0];  // pred_scores      (B,L,1)
  const float* pd     = (const float*)d_in[1];  // pred_distri      (B,L,68)
  const float* ppl    = (const float*)d_in[2];  // pred_pose_logits (B,L,17,3)
  const float* ap     = (const float*)d_in[3];  // anchor_points    (L,2)
  const float* st     = (const float*)d_in[4];  // stride_tensor    (L,1)
  const float* gtb    = (const float*)d_in[5];  // gt_bboxes        (B,N,4)
  const float* padm   = (const float*)d_in[6];  // pad_gt_mask      (B,N,1)
  const float* gposes = (const float*)d_in[7];  // gt_poses         (B,N,17,3)
  const float* sigs   = (const float*)d_in[8];  // oks_sigmas       (17,)
  // d_in[9] = gt_class: all zeros with NUM_CLASSES==1 -> unused.

  char* ws = (char*)d_ws;
  size_t off = 0;
  auto carve = [&](size_t bytes) -> char* {
    char* p = ws + off;
    off = (off + bytes + 255) & ~(size_t)255;
    return p;
  };
  float*         pbs      = (float*)carve(sizeof(float) * (size_t)kB * kL * 4);
  float*         metrics  = (float*)carve(sizeof(float) * (size_t)kB * kN * kL);
  unsigned char* mask     = (unsigned char*)carve((size_t)kB * kN * kL);
  float*         mmax     = (float*)carve(sizeof(float) * kB * kN);
  float*         imax     = (float*)carve(sizeof(float) * kB * kN);
  int*           agi      = (int*)carve(sizeof(int) * (size_t)kB * kL);
  unsigned char* posb     = (unsigned char*)carve((size_t)kB * kL);
  const int nb = (kB * kL + 255) / 256;                     // 525 (exact)
  float*         partials = (float*)carve(sizeof(float) * (size_t)nb * 8);

  // K1: decode (WMMA). 33600 waves, 8 waves/block -> 4200 blocks, exact fill.
  ynp_decode<<<dim3((kB * kL / 4) / 8), 256, 0, stream>>>(pd, ap, st, pbs);
  // K2: metrics + mask clear.
  ynp_metric<<<dim3((kL + 255) / 256, kB * kN), 256, 0, stream>>>(
      ps, pbs, st, gtb, metrics, mask);
  // K3: LDS top-k scatter.
  ynp_topk<<<dim3(kB * kN), 256, 0, stream>>>(metrics, ap, gtb, padm, mask);
  // K4: finalize mask / pos / assigned gt.
  ynp_assign<<<dim3(nb), 256, 0, stream>>>(pbs, st, gtb, mask, agi, posb);
  // K5: per-gt maxima.
  ynp_gtmax<<<dim3(kB * kN), 256, 0, stream>>>(metrics, mask, pbs, st, gtb,
                                               mmax, imax);
  // K7: fused losses with deterministic block partials.
  ynp_loss<<<dim3(nb), 256, 0, stream>>>(ps, pd, ppl, ap, st, gtb, gposes,
                                         sigs, pbs, metrics, mask, agi, posb,
                                         mmax, imax, partials);
  // K8: finalize.
  ynp_final<<<dim3(1), 1, 0, stream>>>(partials, nb, (float*)d_out);
}